// Graphcnn_3513283248949
// MI455X (gfx1250) — compile-verified
//
#include <hip/hip_runtime.h>
#include <math.h>

// ---------------- CDNA5 WMMA (fp32, 16x16x4) ----------------
typedef __attribute__((ext_vector_type(2))) float v2f;
typedef __attribute__((ext_vector_type(8))) float v8f;

#define BB 8
#define NN 2048
#define KK 20
#define EPSF 1e-5f

__device__ __forceinline__ v8f wmma_f32(v2f a, v2f b, v8f c) {
  return __builtin_amdgcn_wmma_f32_16x16x4_f32(false, a, false, b, (short)0, c,
                                               false, false);
}

// branch-free masked load: clamp address into range, select value.
__device__ __forceinline__ float ldm(const float* p, int c, int C) {
  int cc = (c < C) ? c : (C - 1);
  float v = p[(long)cc * NN];
  return (c < C) ? v : 0.f;
}
// masked weight read
__device__ __forceinline__ float wdm(const float* wr, int c, int twoC) {
  float w = wr[(c < twoC) ? c : 0];
  return (c < twoC) ? w : 0.f;
}
// masked edge value: [neigh-ctr ; ctr ; 0-pad]
__device__ __forceinline__ float edm(const float* np, const float* cp, int c,
                                     int C, int twoC) {
  int cn = (c < C) ? c : (C - 1);
  int cc = c - C; cc = (cc < 0) ? 0 : ((cc < C) ? cc : (C - 1));
  float vn = np[(long)cn * NN] - cp[(long)cn * NN];
  float vc = cp[(long)cc * NN];
  float v = (c < C) ? vn : vc;
  return (c < twoC) ? v : 0.f;
}

// ---------------- squared norms: xx[b][n] = sum_c x^2 ----------------
__global__ void k_sqnorm(const float* __restrict__ x, long bstride, int C,
                         float* __restrict__ xx) {
  int t = blockIdx.x * blockDim.x + threadIdx.x;
  if (t >= BB * NN) return;
  int b = t / NN, n = t % NN;
  const float* p = x + (long)b * bstride + n;
  float s = 0.f;
  for (int c = 0; c < C; ++c) { float v = p[(long)c * NN]; s += v * v; }
  xx[t] = s;
}

// ---------------- Gram + pairwise distance via WMMA (32x32/wave) -----------
// pd[b][n][m] = 2*<x_n,x_m> - xx[n] - xx[m]
__global__ void k_gram_pd(const float* __restrict__ x, long bstride, int C,
                          const float* __restrict__ xx, float* __restrict__ pd) {
  const int TPD = NN / 32;              // 64 tiles per dim
  const int TPB = TPD * TPD;
  int lane = threadIdx.x;
  long tile = (long)blockIdx.x * blockDim.y + threadIdx.y;
  if (tile >= (long)BB * TPB) return;
  int b = (int)(tile / TPB);
  int rem = (int)(tile % TPB);
  int rowBase = (rem / TPD) * 32;
  int colBase = (rem % TPD) * 32;
  const float* xb = x + (long)b * bstride;
  int m = lane & 15;
  int sel = (lane >> 4) * 2;
  const float* pr0 = xb + rowBase + m;
  const float* pr1 = pr0 + 16;
  const float* pc0 = xb + colBase + m;
  const float* pc1 = pc0 + 16;
  v8f a00 = {0,0,0,0,0,0,0,0}, a01 = a00, a10 = a00, a11 = a00;
  int C4 = C & ~3;
  for (int k0 = 0; k0 < C4; k0 += 4) {
    long c0 = k0 + sel, c1 = c0 + 1;
    v2f A0, A1, B0, B1;
    A0.x = pr0[c0 * NN]; A0.y = pr0[c1 * NN];
    A1.x = pr1[c0 * NN]; A1.y = pr1[c1 * NN];
    B0.x = pc0[c0 * NN]; B0.y = pc0[c1 * NN];
    B1.x = pc1[c0 * NN]; B1.y = pc1[c1 * NN];
    a00 = wmma_f32(A0, B0, a00); a01 = wmma_f32(A0, B1, a01);
    a10 = wmma_f32(A1, B0, a10); a11 = wmma_f32(A1, B1, a11);
  }
  if (C4 < C) {  // branch-free masked tail (C = 3 case)
    int c0 = C4 + sel, c1 = c0 + 1;
    v2f A0, A1, B0, B1;
    A0.x = ldm(pr0, c0, C); A0.y = ldm(pr0, c1, C);
    A1.x = ldm(pr1, c0, C); A1.y = ldm(pr1, c1, C);
    B0.x = ldm(pc0, c0, C); B0.y = ldm(pc0, c1, C);
    B1.x = ldm(pc1, c0, C); B1.y = ldm(pc1, c1, C);
    a00 = wmma_f32(A0, B0, a00); a01 = wmma_f32(A0, B1, a01);
    a10 = wmma_f32(A1, B0, a10); a11 = wmma_f32(A1, B1, a11);
  }
  int col0 = colBase + m, col1 = col0 + 16;
  int r0 = rowBase + ((lane < 16) ? 0 : 8);
  const float* xxb = xx + b * NN;
  float xc0 = xxb[col0], xc1 = xxb[col1];
  float* pdb = pd + (long)b * NN * NN;
#pragma unroll
  for (int r = 0; r < 8; ++r) {
    int rowA = r0 + r, rowB = r0 + 16 + r;
    float xrA = xxb[rowA], xrB = xxb[rowB];
    pdb[(long)rowA * NN + col0] = 2.f * a00[r] - xrA - xc0;
    pdb[(long)rowA * NN + col1] = 2.f * a01[r] - xrA - xc1;
    pdb[(long)rowB * NN + col0] = 2.f * a10[r] - xrB - xc0;
    pdb[(long)rowB * NN + col1] = 2.f * a11[r] - xrB - xc1;
  }
}

// ---------------- row-min + masked top-20 (register insertion sort) --------
__global__ void k_knn(const float* __restrict__ pd, const int* __restrict__ mask,
                      int invert, int* __restrict__ idx) {
  int t = blockIdx.x * blockDim.x + threadIdx.x;  // b*N+n
  if (t >= BB * NN) return;
  int b = t / NN;
  const float* row = pd + (long)t * NN;
  const int* mb = mask + b * NN;
  float mn = 3.4e38f;
  for (int mcol = 0; mcol < NN; ++mcol) mn = fminf(mn, row[mcol]);
  float bv[KK]; int bi[KK];
#pragma unroll
  for (int j = 0; j < KK; ++j) { bv[j] = -3.4e38f; bi[j] = 0; }
  for (int mcol = 0; mcol < NN; ++mcol) {
    int keep = (mb[mcol] != 0);
    if (invert) keep = !keep;
    float v = keep ? row[mcol] : mn;
    if (v > bv[KK - 1]) {
#pragma unroll
      for (int j = KK - 1; j >= 1; --j) {
        if (v > bv[j - 1])      { bv[j] = bv[j - 1]; bi[j] = bi[j - 1]; }
        else if (v > bv[j])     { bv[j] = v;         bi[j] = mcol; }
      }
      if (v > bv[0]) { bv[0] = v; bi[0] = mcol; }
    }
  }
  int* ob = idx + (long)t * KK;
#pragma unroll
  for (int j = 0; j < KK; ++j) ob[j] = bi[j];
}

// ---------------- graph_layer attention (per point) ----------------
__global__ void k_attn(const float* __restrict__ feat, long bstride, int C,
                       const int* __restrict__ idx, float* __restrict__ out) {
  __shared__ float kx[KK * 128];
  __shared__ float at[KK * KK];
  __shared__ float sc[KK];
  int pnt = blockIdx.x;  // b*N+n
  int b = pnt / NN, n = pnt % NN;
  int tid = threadIdx.x;  // 0..127
  const int* ib = idx + (long)pnt * KK;
  const float* fb = feat + (long)b * bstride;
  for (int t = tid; t < KK * C; t += 128) {
    int k = t / C, c = t % C;
    kx[t] = fb[(long)c * NN + ib[k]];
  }
  __syncthreads();
  float scale = rsqrtf((float)C);
  for (int p = tid; p < KK * KK; p += 128) {
    int k = p / KK, j = p % KK;
    float d = 0.f;
    for (int c = 0; c < C; ++c) d += kx[k * C + c] * kx[j * C + c];
    at[p] = d * scale;
  }
  __syncthreads();
  if (tid < KK) {
    float mx = -3.4e38f;
    for (int j = 0; j < KK; ++j) mx = fmaxf(mx, at[tid * KK + j]);
    float s = 0.f;
    for (int j = 0; j < KK; ++j) {
      float e = __expf(at[tid * KK + j] - mx);
      at[tid * KK + j] = e; s += e;
    }
    float inv = 1.f / s;
    for (int j = 0; j < KK; ++j) at[tid * KK + j] *= inv;
  }
  __syncthreads();
  if (tid < KK) {
    float s = 0.f;
    for (int k = 0; k < KK; ++k) s += at[k * KK + tid];
    sc[tid] = s;
  }
  __syncthreads();
  for (int c = tid; c < C; c += 128) {
    float o = 0.f;
    for (int j = 0; j < KK; ++j) o += sc[j] * kx[j * C + c];
    out[((long)b * C + c) * NN + n] = o * (1.f / KK);
  }
}

// ---------------- edge conv1x1 via WMMA (32 ch x 32 cols / wave) -----------
// f[:,col] = [neigh(:,j) - ctr(:,n) ; ctr(:,n)] built on the fly.
__global__ void k_conv_edge(const float* __restrict__ neigh, long nbs,
                            const float* __restrict__ ctr, long cbs,
                            const int* __restrict__ idx, int C,
                            const float* __restrict__ W, float* __restrict__ y) {
  const long NCOLS = (long)BB * NN * KK;
  int lane = threadIdx.x;
  int m = lane & 15;
  int sel = (lane >> 4) * 2;
  long colTile = (long)blockIdx.x * blockDim.y + threadIdx.y;
  long col0 = colTile * 32 + m;
  long col1 = col0 + 16;
  // decode both columns
  int kk0 = (int)(col0 % KK); long bn0 = col0 / KK;
  int n0 = (int)(bn0 % NN);   int bb0 = (int)(bn0 / NN);
  int j0 = idx[bn0 * KK + kk0];
  int kk1 = (int)(col1 % KK); long bn1 = col1 / KK;
  int n1 = (int)(bn1 % NN);   int bb1 = (int)(bn1 / NN);
  int j1 = idx[bn1 * KK + kk1];
  const float* np0 = neigh + (long)bb0 * nbs + j0;
  const float* cp0 = ctr + (long)bb0 * cbs + n0;
  const float* np1 = neigh + (long)bb1 * nbs + j1;
  const float* cp1 = ctr + (long)bb1 * cbs + n1;
  int rowBase = blockIdx.y * 32;
  int twoC = 2 * C;
  const float* wr0 = W + (long)(rowBase + m) * twoC;
  const float* wr1 = wr0 + (long)16 * twoC;
  v8f a00 = {0,0,0,0,0,0,0,0}, a01 = a00, a10 = a00, a11 = a00;
  if ((C & 3) == 0) {
    // phase 1: c in [0,C) -> neigh - ctr (no guards)
    for (int k0 = 0; k0 < C; k0 += 4) {
      long c0 = k0 + sel, c1 = c0 + 1;
      v2f A0, A1, B0, B1;
      A0.x = wr0[c0]; A0.y = wr0[c1];
      A1.x = wr1[c0]; A1.y = wr1[c1];
      B0.x = np0[c0 * NN] - cp0[c0 * NN];
      B0.y = np0[c1 * NN] - cp0[c1 * NN];
      B1.x = np1[c0 * NN] - cp1[c0 * NN];
      B1.y = np1[c1 * NN] - cp1[c1 * NN];
      a00 = wmma_f32(A0, B0, a00); a01 = wmma_f32(A0, B1, a01);
      a10 = wmma_f32(A1, B0, a10); a11 = wmma_f32(A1, B1, a11);
    }
    // phase 2: c in [C,2C) -> ctr (no guards)
    for (int k0 = C; k0 < twoC; k0 += 4) {
      long c0 = k0 + sel, cc = c0 - C;
      v2f A0, A1, B0, B1;
      A0.x = wr0[c0]; A0.y = wr0[c0 + 1];
      A1.x = wr1[c0]; A1.y = wr1[c0 + 1];
      B0.x = cp0[cc * NN]; B0.y = cp0[(cc + 1) * NN];
      B1.x = cp1[cc * NN]; B1.y = cp1[(cc + 1) * NN];
      a00 = wmma_f32(A0, B0, a00); a01 = wmma_f32(A0, B1, a01);
      a10 = wmma_f32(A1, B0, a10); a11 = wmma_f32(A1, B1, a11);
    }
  } else {
    // generic branch-free masked path (C=3, twoC=6)
    for (int k0 = 0; k0 < twoC; k0 += 4) {
      int c0 = k0 + sel, c1 = c0 + 1;
      v2f A0, A1, B0, B1;
      A0.x = wdm(wr0, c0, twoC); A0.y = wdm(wr0, c1, twoC);
      A1.x = wdm(wr1, c0, twoC); A1.y = wdm(wr1, c1, twoC);
      B0.x = edm(np0, cp0, c0, C, twoC); B0.y = edm(np0, cp0, c1, C, twoC);
      B1.x = edm(np1, cp1, c0, C, twoC); B1.y = edm(np1, cp1, c1, C, twoC);
      a00 = wmma_f32(A0, B0, a00); a01 = wmma_f32(A0, B1, a01);
      a10 = wmma_f32(A1, B0, a10); a11 = wmma_f32(A1, B1, a11);
    }
  }
  int r0 = rowBase + ((lane < 16) ? 0 : 8);
#pragma unroll
  for (int r = 0; r < 8; ++r) {
    long rowA = r0 + r, rowB = r0 + 16 + r;
    y[rowA * NCOLS + col0] = a00[r];
    y[rowA * NCOLS + col1] = a01[r];
    y[rowB * NCOLS + col0] = a10[r];
    y[rowB * NCOLS + col1] = a11[r];
  }
}

// ---------------- point conv (W9) with mask select (32x32/wave) ------------
__global__ void k_conv_point(const float* __restrict__ xl,
                             const float* __restrict__ xg,
                             const int* __restrict__ mask,
                             const float* __restrict__ W,  // 1024 x 512
                             float* __restrict__ y) {
  const int CIN = 512;
  const long NCOLS = (long)BB * NN;
  int lane = threadIdx.x;
  int m = lane & 15, sel = (lane >> 4) * 2;
  long colTile = (long)blockIdx.x * blockDim.y + threadIdx.y;
  long col0 = colTile * 32 + m;
  long col1 = col0 + 16;
  int n0 = (int)(col0 % NN), b0 = (int)(col0 / NN);
  int n1 = (int)(col1 % NN), b1 = (int)(col1 / NN);
  const float* s0 = (mask[b0 * NN + n0] != 0) ? xl : xg;
  const float* s1 = (mask[b1 * NN + n1] != 0) ? xl : xg;
  const float* p0 = s0 + (long)b0 * CIN * NN + n0;
  const float* p1 = s1 + (long)b1 * CIN * NN + n1;
  int rowBase = blockIdx.y * 32;
  const float* wr0 = W + (long)(rowBase + m) * CIN;
  const float* wr1 = wr0 + (long)16 * CIN;
  v8f a00 = {0,0,0,0,0,0,0,0}, a01 = a00, a10 = a00, a11 = a00;
  for (int k0 = 0; k0 < CIN; k0 += 4) {
    long c0 = k0 + sel, c1 = c0 + 1;
    v2f A0, A1, B0, B1;
    A0.x = wr0[c0]; A0.y = wr0[c1];
    A1.x = wr1[c0]; A1.y = wr1[c1];
    B0.x = p0[c0 * NN]; B0.y = p0[c1 * NN];
    B1.x = p1[c0 * NN]; B1.y = p1[c1 * NN];
    a00 = wmma_f32(A0, B0, a00); a01 = wmma_f32(A0, B1, a01);
    a10 = wmma_f32(A1, B0, a10); a11 = wmma_f32(A1, B1, a11);
  }
  int r0 = rowBase + ((lane < 16) ? 0 : 8);
#pragma unroll
  for (int r = 0; r < 8; ++r) {
    long rowA = r0 + r, rowB = r0 + 16 + r;
    y[rowA * NCOLS + col0] = a00[r];
    y[rowA * NCOLS + col1] = a01[r];
    y[rowB * NCOLS + col0] = a10[r];
    y[rowB * NCOLS + col1] = a11[r];
  }
}

// ---------------- per-channel mean/var over y row ----------------
__global__ void k_stats(const float* __restrict__ y, long ncols,
                        float* __restrict__ mv) {
  __shared__ float s1[256], s2[256];
  int o = blockIdx.x;
  const float* p = y + (long)o * ncols;
  float a = 0.f, b = 0.f;
  for (long i = threadIdx.x; i < ncols; i += 256) {
    float v = p[i]; a += v; b += v * v;
  }
  s1[threadIdx.x] = a; s2[threadIdx.x] = b;
  __syncthreads();
  for (int st = 128; st > 0; st >>= 1) {
    if ((int)threadIdx.x < st) {
      s1[threadIdx.x] += s1[threadIdx.x + st];
      s2[threadIdx.x] += s2[threadIdx.x + st];
    }
    __syncthreads();
  }
  if (threadIdx.x == 0) {
    float mean = s1[0] / (float)ncols;
    float var = s2[0] / (float)ncols - mean * mean;
    mv[2 * o] = mean; mv[2 * o + 1] = var;
  }
}

// ---------------- bn (+opt lrelu) + max over k, write (B,totC,N) ----------
__global__ void k_bnmax(const float* __restrict__ y, int O,
                        const float* __restrict__ mv,
                        const float* __restrict__ g, const float* __restrict__ bt,
                        int lrelu, float* __restrict__ out, int chOff, int totC) {
  long t = (long)blockIdx.x * blockDim.x + threadIdx.x;
  long total = (long)BB * O * NN;
  if (t >= total) return;
  int n = (int)(t % NN);
  long bo = t / NN;
  int o = (int)(bo % O);
  int b = (int)(bo / O);
  float mean = mv[2 * o], var = mv[2 * o + 1];
  float sc = g[o] * rsqrtf(var + EPSF);
  float sh = bt[o] - mean * sc;
  const long NCOLS = (long)BB * NN * KK;
  const float* p = y + (long)o * NCOLS + ((long)b * NN + n) * KK;
  float mx = -3.4e38f;
#pragma unroll 4
  for (int k = 0; k < KK; ++k) {
    float v = sc * p[k] + sh;
    if (lrelu && v < 0.f) v *= 0.2f;
    mx = fmaxf(mx, v);
  }
  out[((long)b * totC + chOff + o) * NN + n] = mx;
}

// ---------------- bn9 + lrelu + max&mean over N -> h(8,2048) ----------------
__global__ void k_bn9_maxmean(const float* __restrict__ y,
                              const float* __restrict__ mv,
                              const float* __restrict__ g,
                              const float* __restrict__ bt,
                              float* __restrict__ h) {
  int t = blockIdx.x * blockDim.x + threadIdx.x;  // b*1024+o
  if (t >= BB * 1024) return;
  int b = t / 1024, o = t % 1024;
  float mean = mv[2 * o], var = mv[2 * o + 1];
  float sc = g[o] * rsqrtf(var + EPSF), sh = bt[o] - mean * sc;
  const float* p = y + (long)o * (BB * NN) + (long)b * NN;
  float mx = -3.4e38f, sm = 0.f;
  for (int n = 0; n < NN; ++n) {
    float v = sc * p[n] + sh;
    if (v < 0.f) v *= 0.2f;
    mx = fmaxf(mx, v); sm += v;
  }
  h[b * 2048 + o] = mx;
  h[b * 2048 + 1024 + o] = sm * (1.f / NN);
}

// ---------------- tiny FC: out(8,Cout) = in(8,Cin) @ W^T + bias -------------
__global__ void k_fc(const float* __restrict__ in, int Cin,
                     const float* __restrict__ W, const float* __restrict__ bias,
                     int Cout, float* __restrict__ out) {
  int t = blockIdx.x * blockDim.x + threadIdx.x;
  if (t >= BB * Cout) return;
  int b = t / Cout, o = t % Cout;
  const float* ip = in + (long)b * Cin;
  const float* wp = W + (long)o * Cin;
  float s = bias ? bias[o] : 0.f;
  for (int c = 0; c < Cin; ++c) s += ip[c] * wp[c];
  out[t] = s;
}

// ---------------- bn over batch axis (8) + lrelu, in place ------------------
__global__ void k_bnf_lrelu(float* __restrict__ x, int C,
                            const float* __restrict__ g,
                            const float* __restrict__ bt) {
  int c = blockIdx.x * blockDim.x + threadIdx.x;
  if (c >= C) return;
  float m = 0.f;
  for (int b = 0; b < BB; ++b) m += x[b * C + c];
  m *= (1.f / BB);
  float v = 0.f;
  for (int b = 0; b < BB; ++b) { float d = x[b * C + c] - m; v += d * d; }
  v *= (1.f / BB);
  float sc = g[c] * rsqrtf(v + EPSF);
  for (int b = 0; b < BB; ++b) {
    float t = sc * (x[b * C + c] - m) + bt[c];
    x[b * C + c] = (t < 0.f) ? 0.2f * t : t;
  }
}

// =========================== host orchestration =============================
extern "C" void kernel_launch(void* const* d_in, const int* in_sizes, int n_in,
                              void* d_out, int out_size, void* d_ws,
                              size_t ws_size, hipStream_t stream) {
  (void)in_sizes; (void)n_in; (void)out_size; (void)ws_size;
  const float* x   = (const float*)d_in[0];
  const int* mask  = (const int*)d_in[1];
  const float* P[36];
  for (int i = 0; i < 36; ++i) P[i] = (const float*)d_in[2 + i];
  const float *W1=P[0],*g1=P[1],*b1=P[2], *W2=P[3],*g2=P[4],*b2=P[5];
  const float *W3=P[6],*g3=P[7],*b3=P[8], *W4=P[9],*g4=P[10],*b4=P[11];
  const float *W5=P[12],*g5=P[13],*b5=P[14], *W6=P[15],*g6=P[16],*b6=P[17];
  const float *W7=P[18],*g7=P[19],*b7=P[20], *W8=P[21],*g8=P[22],*b8=P[23];
  const float *W9=P[24],*g9=P[25],*b9=P[26];
  const float *Wl1=P[27],*g10=P[28],*b10=P[29];
  const float *Wl2=P[30],*bl2=P[31],*g11=P[32],*b11=P[33];
  const float *Wl3=P[34],*bl3=P[35];

  // workspace layout (floats)
  float* ws = (float*)d_ws;
  size_t off = 0;
  float* pd    = ws + off; off += (size_t)BB * NN * NN;        // 33.5M
  float* xx    = ws + off; off += (size_t)BB * NN;
  float* attn  = ws + off; off += (size_t)BB * 128 * NN;       // 2.1M
  float* xloc  = ws + off; off += (size_t)BB * 512 * NN;       // 8.4M
  float* xg    = ws + off; off += (size_t)BB * 512 * NN;       // 8.4M
  float* ybuf  = ws + off; off += (size_t)256 * BB * NN * KK;  // 83.9M
  float* mv    = ws + off; off += 2 * 1024;
  float* hbuf  = ws + off; off += (size_t)BB * 2048;
  float* t1    = ws + off; off += (size_t)BB * 512;
  float* t2    = ws + off; off += (size_t)BB * 256;
  int* idx     = (int*)(ws + off);

  const long BS512 = (long)512 * NN;
  const long NCOLS_E = (long)BB * NN * KK;  // 327680 (= 32*10240)
  dim3 wblk(32, 8);  // 8 waves per WG

  auto knn = [&](const float* feat, long bstride, int C, int invert) {
    k_sqnorm<<<(BB * NN + 255) / 256, 256, 0, stream>>>(feat, bstride, C, xx);
    int tiles = BB * (NN / 32) * (NN / 32);  // 32768
    k_gram_pd<<<(tiles + 7) / 8, wblk, 0, stream>>>(feat, bstride, C, xx, pd);
    k_knn<<<(BB * NN + 127) / 128, 128, 0, stream>>>(pd, mask, invert, idx);
  };
  auto edge_conv = [&](const float* nb, long nbs, const float* cb, long cbs,
                       int C, const float* W, int O, const float* g,
                       const float* bt, int lrelu, float* out, int chOff) {
    dim3 grd((unsigned)(NCOLS_E / 32 / 8), (unsigned)(O / 32));
    k_conv_edge<<<grd, wblk, 0, stream>>>(nb, nbs, cb, cbs, idx, C, W, ybuf);
    k_stats<<<O, 256, 0, stream>>>(ybuf, NCOLS_E, mv);
    long tot = (long)BB * O * NN;
    k_bnmax<<<(unsigned)((tot + 255) / 256), 256, 0, stream>>>(
        ybuf, O, mv, g, bt, lrelu, out, chOff, 512);
  };

  // ---- local branch ----
  knn(x, (long)3 * NN, 3, 0);
  edge_conv(x, (long)3 * NN, x, (long)3 * NN, 3, W1, 64, g1, b1, 1, xloc, 0);
  knn(xloc, BS512, 64, 0);
  k_attn<<<BB * NN, 128, 0, stream>>>(xloc, BS512, 64, idx, attn);
  edge_conv(attn, (long)64 * NN, xloc, BS512, 64, W2, 64, g2, b2, 0, xloc, 64);
  knn(xloc + (long)64 * NN, BS512, 64, 0);
  k_attn<<<BB * NN, 128, 0, stream>>>(xloc + (long)64 * NN, BS512, 64, idx, attn);
  edge_conv(attn, (long)64 * NN, xloc + (long)64 * NN, BS512, 64, W3, 128, g3,
            b3, 0, xloc, 128);
  knn(xloc + (long)128 * NN, BS512, 128, 0);
  k_attn<<<BB * NN, 128, 0, stream>>>(xloc + (long)128 * NN, BS512, 128, idx,
                                      attn);
  edge_conv(attn, (long)128 * NN, xloc + (long)128 * NN, BS512, 128, W4, 256,
            g4, b4, 0, xloc, 256);

  // ---- global branch (inverted mask) ----
  knn(x, (long)3 * NN, 3, 1);
  edge_conv(x, (long)3 * NN, x, (long)3 * NN, 3, W5, 64, g5, b5, 0, xg, 0);
  knn(xg, BS512, 64, 1);
  edge_conv(xg, BS512, xg, BS512, 64, W6, 64, g6, b6, 0, xg, 64);
  knn(xg + (long)64 * NN, BS512, 64, 1);
  edge_conv(xg + (long)64 * NN, BS512, xg + (long)64 * NN, BS512, 64, W7, 128,
            g7, b7, 0, xg, 128);
  knn(xg + (long)128 * NN, BS512, 128, 1);
  edge_conv(xg + (long)128 * NN, BS512, xg + (long)128 * NN, BS512, 128, W8,
            256, g8, b8, 1, xg, 256);

  // ---- head: W9 conv over mask-selected features ----
  k_conv_point<<<dim3(64, 32), wblk, 0, stream>>>(xloc, xg, mask, W9, ybuf);
  k_stats<<<1024, 256, 0, stream>>>(ybuf, (long)BB * NN, mv);
  k_bn9_maxmean<<<(BB * 1024 + 255) / 256, 256, 0, stream>>>(ybuf, mv, g9, b9,
                                                             hbuf);
  // ---- FC stack ----
  k_fc<<<(BB * 512 + 255) / 256, 256, 0, stream>>>(hbuf, 2048, Wl1, nullptr,
                                                   512, t1);
  k_bnf_lrelu<<<(512 + 63) / 64, 64, 0, stream>>>(t1, 512, g10, b10);
  k_fc<<<(BB * 256 + 255) / 256, 256, 0, stream>>>(t1, 512, Wl2, bl2, 256, t2);
  k_bnf_lrelu<<<(256 + 63) / 64, 64, 0, stream>>>(t2, 256, g11, b11);
  k_fc<<<(BB * 40 + 255) / 256, 256, 0, stream>>>(t2, 256, Wl3, bl3, 40,
                                                  (float*)d_out);
}